// MESNReadout_37812892074593
// MI455X (gfx1250) — compile-verified
//
#include <hip/hip_runtime.h>

#define LNUM 3
#define SNUM 4
#define TH   5
#define IND  64
#define NCLS 100
#define BB   512
#define TSTEPS 1024
#define TC   2          // timesteps per chunk (double-buffered)
#define NCHUNK (TSTEPS / TC)
#define NPADR 76        // proj row stride in floats (16B-aligned rows)
#define NCOLS 72        // 3*20 layer-projection cols + 12 xv cols

typedef float v2f __attribute__((ext_vector_type(2)));
typedef float v8f __attribute__((ext_vector_type(8)));

__device__ __forceinline__ float tanh_fast(float x) {
    // tanh(x) = sign(x) * (1 - e^{-2|x|}) / (1 + e^{-2|x|}); stable for all x
    float ax = __builtin_fabsf(x);
    float e  = __builtin_amdgcn_exp2f(ax * -2.8853900817779268f); // exp(-2*ax)
    float t  = (1.0f - e) * __builtin_amdgcn_rcpf(1.0f + e);
    return __builtin_copysignf(t, x);
}

__global__ __launch_bounds__(64)
void mesn_fused_kernel(const float* __restrict__ u,     // [B, T, 64]
                       const float* __restrict__ Wi0,   // [S, 64, TH]
                       const float* __restrict__ Wir,   // [L-1, S, 69, TH]
                       const float* __restrict__ Wl,    // [L, S, TH, TH]
                       const float* __restrict__ Wvin,  // [12, 64]
                       const float* __restrict__ Wv,    // [12, 12]
                       const float* __restrict__ Wout,  // [72, 100]
                       const float* __restrict__ bout,  // [100]
                       float* __restrict__ out)         // [B, 100]
{
    // ---- LDS: 63168 bytes total ----
    __shared__ __align__(16) float s_bfrag[5][16][32][2];        // 20480 B: W_cat in WMMA B-frag layout
    __shared__ __align__(16) float s_proj[2][TC][32][NPADR];     // 38912 B: double-buffered projections
    __shared__ __align__(16) float s_wl[LNUM][SNUM][TH][8];      //  1920 B
    __shared__ __align__(16) float s_wr[LNUM-1][SNUM][TH][8];    //  1280 B
    __shared__ __align__(16) float s_wv[12][12];                 //   576 B

    const int tid  = threadIdx.x;
    const int wave = tid >> 5;       // 0 = GEMM producer, 1 = scan consumer
    const int lane = tid & 31;
    const int b0   = blockIdx.x * 32;
    const int hsel = lane >> 4;      // A/B fragment half (K pair select)
    const int lmod = lane & 15;

    // ================= one-time weight staging (both waves) =================
    // W_cat column n: [0,20)=layer0 W_in0 ; [20,60)=W_in_rest[d][:, :64] ; [60,72)=Wv_in.T ; [72,80)=0
    for (int idx = tid; idx < 5 * 16 * 32 * 2; idx += 64) {
        int v  = idx & 1;
        int l  = (idx >> 1) & 31;
        int kk = (idx >> 6) & 15;
        int nt = idx >> 10;
        int k  = 4 * kk + v + ((l >> 4) << 1);
        int n  = nt * 16 + (l & 15);
        float val = 0.0f;
        if (n < 20) {
            int s = n / 5, j = n % 5;
            val = Wi0[(s * IND + k) * TH + j];
        } else if (n < 60) {
            int d = (n - 20) / 20, r = (n - 20) % 20;
            int s = r / 5, j = r % 5;
            val = Wir[((d * SNUM + s) * (IND + TH) + k) * TH + j];
        } else if (n < NCOLS) {
            val = Wvin[(n - 60) * IND + k];
        }
        s_bfrag[nt][kk][l][v] = val;
    }
    for (int idx = tid; idx < LNUM * SNUM * TH; idx += 64) {
        int d = idx / (SNUM * TH), s = (idx / TH) % SNUM, i = idx % TH;
        for (int j = 0; j < TH; ++j)
            s_wl[d][s][i][j] = Wl[((d * SNUM + s) * TH + i) * TH + j];
    }
    for (int idx = tid; idx < (LNUM - 1) * SNUM * TH; idx += 64) {
        int d = idx / (SNUM * TH), s = (idx / TH) % SNUM, i = idx % TH;
        for (int j = 0; j < TH; ++j)
            s_wr[d][s][i][j] = Wir[((d * SNUM + s) * (IND + TH) + IND + i) * TH + j];
    }
    if (tid < 12)
        for (int i = 0; i < 12; ++i) s_wv[tid][i] = Wv[tid * 12 + i];
    __syncthreads();

    // ================= producer state: B fragments held in VGPRs =================
    v2f bf[5][16];
    // ================= consumer state: lane = batch row b0+lane =================
    float x[LNUM][SNUM][TH];
    float xv[12];
#pragma unroll
    for (int d = 0; d < LNUM; ++d)
#pragma unroll
        for (int s = 0; s < SNUM; ++s)
#pragma unroll
            for (int i = 0; i < TH; ++i) x[d][s][i] = 0.0f;
#pragma unroll
    for (int c = 0; c < 12; ++c) xv[c] = 0.0f;

    // -------- producer: projection GEMM for one chunk into buffer `buf` --------
    auto gemm_chunk = [&](int t0, int buf) {
#pragma unroll 1
        for (int tt = 0; tt < TC; ++tt) {
#pragma unroll
            for (int h = 0; h < 2; ++h) {
                const float* arow =
                    u + ((size_t)(b0 + 16 * h + lmod) * TSTEPS + (t0 + tt)) * IND + 2 * hsel;
                v8f acc0 = {0,0,0,0,0,0,0,0};
                v8f acc1 = acc0, acc2 = acc0, acc3 = acc0, acc4 = acc0;
#pragma unroll
                for (int kk = 0; kk < 16; ++kk) {
                    v2f a = *(const v2f*)(arow + 4 * kk);
                    acc0 = __builtin_amdgcn_wmma_f32_16x16x4_f32(
                        false, a, false, bf[0][kk], (short)0, acc0, false, false);
                    acc1 = __builtin_amdgcn_wmma_f32_16x16x4_f32(
                        false, a, false, bf[1][kk], (short)0, acc1, false, false);
                    acc2 = __builtin_amdgcn_wmma_f32_16x16x4_f32(
                        false, a, false, bf[2][kk], (short)0, acc2, false, false);
                    acc3 = __builtin_amdgcn_wmma_f32_16x16x4_f32(
                        false, a, false, bf[3][kk], (short)0, acc3, false, false);
                    acc4 = __builtin_amdgcn_wmma_f32_16x16x4_f32(
                        false, a, false, bf[4][kk], (short)0, acc4, false, false);
                }
                // scatter D fragments: vgpr r -> row (16h + r + 8*hsel), col (16*nt + lmod)
#pragma unroll
                for (int r = 0; r < 8; ++r) {
                    int brow = 16 * h + r + 8 * hsel;
                    s_proj[buf][tt][brow][ 0 + lmod] = acc0[r];
                    s_proj[buf][tt][brow][16 + lmod] = acc1[r];
                    s_proj[buf][tt][brow][32 + lmod] = acc2[r];
                    s_proj[buf][tt][brow][48 + lmod] = acc3[r];
                    if (64 + lmod < NPADR) s_proj[buf][tt][brow][64 + lmod] = acc4[r];
                }
            }
        }
    };

    // -------- consumer: TC sequential recurrence steps from buffer `buf` --------
    auto scan_chunk = [&](int buf) {
#pragma unroll 1
        for (int tt = 0; tt < TC; ++tt) {
            const float* pr = &s_proj[buf][tt][lane][0];
            float p[NCOLS];
#pragma unroll
            for (int f = 0; f < NCOLS; ++f) p[f] = pr[f];

            float rep[12];
#pragma unroll
            for (int d = 0; d < LNUM; ++d) {
#pragma unroll
                for (int s = 0; s < SNUM; ++s) {
                    float a[TH];
#pragma unroll
                    for (int j = 0; j < TH; ++j) a[j] = p[d * 20 + s * TH + j];
#pragma unroll
                    for (int i = 0; i < TH; ++i) {      // x_old[d][s] @ W[d][s]
                        float xi = x[d][s][i];
#pragma unroll
                        for (int j = 0; j < TH; ++j) a[j] += xi * s_wl[d][s][i][j];
                    }
                    if (d > 0) {                        // new x[d-1] @ Win[:,64:]
#pragma unroll
                        for (int i = 0; i < TH; ++i) {
                            float xi = x[d - 1][s][i];
#pragma unroll
                            for (int j = 0; j < TH; ++j) a[j] += xi * s_wr[d - 1][s][i][j];
                        }
                    }
                    float sum = 0.0f;
#pragma unroll
                    for (int j = 0; j < TH; ++j) {
                        float nx = tanh_fast(a[j]);
                        x[d][s][j] = nx;
                        sum += nx;
                    }
                    rep[d * SNUM + s] = sum * 0.2f;
                }
            }
            float tv[12];
#pragma unroll
            for (int c = 0; c < 12; ++c) {
                float a = p[60 + c];
#pragma unroll
                for (int i = 0; i < 12; ++i) a += xv[i] * s_wv[c][i];
                tv[c] = tanh_fast(a);
            }
#pragma unroll
            for (int c = 0; c < 12; ++c) xv[c] = 0.1f * rep[c] + 0.9f * tv[c];
        }
    };

    // ================= pipelined main loop =================
    if (wave == 0) {
        // hoist all B fragments into registers (producer owns its SIMD's RF)
#pragma unroll
        for (int nt = 0; nt < 5; ++nt)
#pragma unroll
            for (int kk = 0; kk < 16; ++kk)
                bf[nt][kk] = *(const v2f*)&s_bfrag[nt][kk][lane][0];
        gemm_chunk(0, 0);                       // prologue: chunk 0
    }
    __syncthreads();

#pragma unroll 1
    for (int c = 0; c < NCHUNK; ++c) {
        if (wave == 0) {
            if (c + 1 < NCHUNK) {
                if (c + 2 < NCHUNK)             // prefetch chunk c+2's u rows
                    __builtin_prefetch(
                        u + ((size_t)(b0 + lane) * TSTEPS + (size_t)(c + 2) * TC) * IND, 0, 1);
                gemm_chunk((c + 1) * TC, (c + 1) & 1);   // produce chunk c+1
            }
        } else {
            scan_chunk(c & 1);                  // consume chunk c
        }
        __syncthreads();
    }

    // ================= readout: out[b] = feats @ W_out + b_out =================
    float* wout = &s_proj[0][0][0][0];          // reuse proj LDS: 7200 floats of W_out
    for (int idx = tid; idx < NCOLS * NCLS; idx += 64) wout[idx] = Wout[idx];
    float* fl = &s_proj[0][0][0][0] + 7232 + lane * 73;  // per-lane feats, odd stride
    if (wave == 1) {
#pragma unroll
        for (int d = 0; d < LNUM; ++d)
#pragma unroll
            for (int s = 0; s < SNUM; ++s)
#pragma unroll
                for (int j = 0; j < TH; ++j) fl[(d * SNUM + s) * TH + j] = x[d][s][j];
#pragma unroll
        for (int c = 0; c < 12; ++c) fl[60 + c] = xv[c];
    }
    __syncthreads();

    if (wave == 1) {
        float acc[NCLS];
#pragma unroll
        for (int c = 0; c < NCLS; ++c) acc[c] = bout[c];
#pragma unroll 1
        for (int f = 0; f < NCOLS; ++f) {
            float v = fl[f];
            const float* wr = wout + f * NCLS;
#pragma unroll
            for (int c = 0; c < NCLS; ++c) acc[c] += v * wr[c];
        }
        float* orow = out + (size_t)(b0 + lane) * NCLS;
#pragma unroll
        for (int c = 0; c < NCLS; ++c) orow[c] = acc[c];
    }
}

extern "C" void kernel_launch(void* const* d_in, const int* in_sizes, int n_in,
                              void* d_out, int out_size, void* d_ws, size_t ws_size,
                              hipStream_t stream) {
    (void)in_sizes; (void)n_in; (void)d_ws; (void)ws_size; (void)out_size;
    const float* u    = (const float*)d_in[0];
    const float* Wi0  = (const float*)d_in[1];
    const float* Wir  = (const float*)d_in[2];
    const float* Wl   = (const float*)d_in[3];
    const float* Wvin = (const float*)d_in[4];
    const float* Wv   = (const float*)d_in[5];
    const float* Wout = (const float*)d_in[6];
    const float* bout = (const float*)d_in[7];
    mesn_fused_kernel<<<dim3(BB / 32), dim3(64), 0, stream>>>(
        u, Wi0, Wir, Wl, Wvin, Wv, Wout, bout, (float*)d_out);
}